// YOLOLoss_76106820485565
// MI455X (gfx1250) — compile-verified
//
#include <hip/hip_runtime.h>
#include <math.h>

// ---------------------------------------------------------------------------
// YOLO loss, sparse-target reformulation for MI455X (gfx1250, wave32).
//
// obj_loss   = S - sum_mask(o)                         S = sum_all softplus(o)
// noobj_loss = 0.5*( S - sum_mask softplus(o) + M*ln2 )
// cls_loss   = (B*G*G - M)*C*ln2 + sum_maskcells[ sum_c sp(p_c) - sum_{set} p_c ]
// coord_loss = 5 * sum_maskcells sum_ch (reg - reg_tgt)^2
//
// Only obj_pred (2MB) is streamed densely; cls/reg are gathered at <=1024 cells.
// Wave reductions use V_WMMA_F32_16X16X4_F32 (B = ones => row sums), final
// cross-half combine via v_readlane (constant lanes 0 and 16) - no LDS.
// ---------------------------------------------------------------------------

typedef __attribute__((ext_vector_type(2))) float v2f;
typedef __attribute__((ext_vector_type(8))) float v8f;

#define B_ 32
#define C_ 80
#define G_ 128
#define N_ 32

__device__ __forceinline__ float softplusf(float x) {
  // stable: max(x,0) + log1p(exp(-|x|)), matches jax.nn.softplus
  return fmaxf(x, 0.0f) + log1pf(expf(-fabsf(x)));
}

// Sum (a0 + a1) across all 32 lanes using one WMMA (A 16x4 f32, B = ones 4x16).
// Must be called from a full wave (EXEC all ones).
__device__ __forceinline__ float wave_sum2(float a0, float a1) {
  v2f A;  A[0] = a0;  A[1] = a1;
  v2f Bm; Bm[0] = 1.0f; Bm[1] = 1.0f;
  v8f Cm = {};
  v8f D = __builtin_amdgcn_wmma_f32_16x16x4_f32(
      /*neg_a=*/false, A, /*neg_b=*/false, Bm,
      /*c_mod=*/(short)0, Cm, /*reuse_a=*/false, /*reuse_b=*/false);
  float s = D[0] + D[1] + D[2] + D[3] + D[4] + D[5] + D[6] + D[7];
  // lanes 0-15 hold sum of rows 0-7, lanes 16-31 hold sum of rows 8-15
  int lo = __builtin_amdgcn_readlane(__float_as_int(s), 0);
  int hi = __builtin_amdgcn_readlane(__float_as_int(s), 16);
  return __int_as_float(lo) + __int_as_float(hi);
}

// ws accumulator slots
// [0]=S(bg softplus)  [1]=sum_mask o  [2]=sum_mask sp(o)  [3]=M
// [4]=cls correction  [5]=coord sum
__global__ void yolo_init(float* acc) {
  if (threadIdx.x < 8) acc[threadIdx.x] = 0.0f;
}

__global__ void yolo_obj_bg(const float* __restrict__ obj, int n4,
                            float* __restrict__ acc) {
  const float4* p = (const float4*)obj;
  float a0 = 0.0f, a1 = 0.0f;
  for (int i = blockIdx.x * blockDim.x + threadIdx.x; i < n4;
       i += gridDim.x * blockDim.x) {
    float4 v = p[i];
    a0 += softplusf(v.x) + softplusf(v.z);
    a1 += softplusf(v.y) + softplusf(v.w);
  }
  float r = wave_sum2(a0, a1);
  if ((threadIdx.x & 31) == 0) atomicAdd(&acc[0], r);
}

__device__ __forceinline__ void cell_of(const float* boxb, int j,
                                        int& gx, int& gy, bool& valid) {
  float x = boxb[j * 4 + 0];
  float y = boxb[j * 4 + 1];
  int fx = (int)floorf(x * (float)G_);
  int fy = (int)floorf(y * (float)G_);
  valid = (fx >= 0) && (fx < G_) && (fy >= 0) && (fy < G_);
  gx = min(max(fx, 0), G_ - 1);
  gy = min(max(fy, 0), G_ - 1);
}

// one block (one wave) per batch, one lane per target
__global__ void yolo_mask(const float* __restrict__ cls_pred,
                          const float* __restrict__ reg_pred,
                          const float* __restrict__ obj_pred,
                          const int* __restrict__ tcls,
                          const float* __restrict__ tbox,
                          float* __restrict__ acc) {
  int b = blockIdx.x;
  int n = threadIdx.x;  // 0..31
  const float* boxb = tbox + (size_t)b * N_ * 4;
  const int* clsb = tcls + b * N_;

  int gx, gy; bool valid;
  cell_of(boxb, n, gx, gy, valid);
  int cell = gy * G_ + gx;
  int mycls = clsb[n];

  // canonical-per-cell: first valid target with this cell
  // canonical-per-(cell,class): first valid target with this (cell,class)
  bool canon_cell = valid;
  bool canon_cc = valid;
  for (int j = 0; j < n; ++j) {
    int jx, jy; bool jv;
    cell_of(boxb, j, jx, jy, jv);
    if (jv && (jy * G_ + jx) == cell) {
      canon_cell = false;
      if (clsb[j] == mycls) canon_cc = false;
    }
  }

  float a_o = 0.0f, a_sp = 0.0f, a_m = 0.0f, a_cls = 0.0f, a_coord = 0.0f;
  size_t spat = (size_t)gy * G_ + gx;

  if (canon_cell) {
    float o = obj_pred[(size_t)b * G_ * G_ + spat];
    a_o = o;
    a_sp = softplusf(o);
    a_m = 1.0f;

    // per-cell cls softplus sum over all 80 channels
    const float* cp = cls_pred + (size_t)b * C_ * G_ * G_ + spat;
    float s = 0.0f;
    for (int c = 0; c < C_; ++c) s += softplusf(cp[(size_t)c * G_ * G_]);
    a_cls = s;

    // reg target: "last writer wins" among targets sharing the cell
    int lastj = n;
    for (int j = n + 1; j < N_; ++j) {
      int jx, jy; bool jv;
      cell_of(boxb, j, jx, jy, jv);
      if (jv && (jy * G_ + jx) == cell) lastj = j;
    }
    float x = boxb[lastj * 4 + 0], y = boxb[lastj * 4 + 1];
    float w = boxb[lastj * 4 + 2], h = boxb[lastj * 4 + 3];
    float rt0 = x * (float)G_ - (float)gx;
    float rt1 = y * (float)G_ - (float)gy;
    float rt2 = sqrtf(w);
    float rt3 = sqrtf(h);
    const float* rp = reg_pred + (size_t)b * 4 * G_ * G_ + spat;
    float d0 = rp[0 * (size_t)G_ * G_] - rt0;
    float d1 = rp[1 * (size_t)G_ * G_] - rt1;
    float d2 = rp[2 * (size_t)G_ * G_] - rt2;
    float d3 = rp[3 * (size_t)G_ * G_] - rt3;
    a_coord = d0 * d0 + d1 * d1 + d2 * d2 + d3 * d3;
  }
  if (canon_cc) {
    // cls_target has a 1 at (cell, mycls): subtract that logit once
    a_cls -= cls_pred[(size_t)b * C_ * G_ * G_ + (size_t)mycls * G_ * G_ + spat];
  }

  // full-wave WMMA reductions (uniform control flow)
  float r_o     = wave_sum2(a_o, 0.0f);
  float r_sp    = wave_sum2(a_sp, 0.0f);
  float r_m     = wave_sum2(a_m, 0.0f);
  float r_cls   = wave_sum2(a_cls, 0.0f);
  float r_coord = wave_sum2(a_coord, 0.0f);
  if (threadIdx.x == 0) {
    atomicAdd(&acc[1], r_o);
    atomicAdd(&acc[2], r_sp);
    atomicAdd(&acc[3], r_m);
    atomicAdd(&acc[4], r_cls);
    atomicAdd(&acc[5], r_coord);
  }
}

__global__ void yolo_finalize(const float* __restrict__ acc,
                              float* __restrict__ out) {
  if (threadIdx.x != 0 || blockIdx.x != 0) return;
  float S = acc[0], Ao = acc[1], Asp = acc[2], M = acc[3];
  float clsS = acc[4], coordS = acc[5];
  const float LN2 = 0.69314718055994530942f;

  float obj_loss = S - Ao;
  float noobj_loss = 0.5f * (S - Asp + M * LN2);
  double bg = ((double)B_ * (double)G_ * (double)G_ - (double)M) *
              (double)C_ * 0.6931471805599453;
  float cls_loss = (float)((double)clsS + bg);
  float coord_loss = 5.0f * coordS;
  float total = obj_loss + noobj_loss + coord_loss + cls_loss;

  const float invB = 1.0f / (float)B_;
  out[0] = total * invB;
  out[1] = obj_loss * invB;
  out[2] = noobj_loss * invB;
  out[3] = coord_loss * invB;
  out[4] = cls_loss * invB;
}

extern "C" void kernel_launch(void* const* d_in, const int* in_sizes, int n_in,
                              void* d_out, int out_size, void* d_ws, size_t ws_size,
                              hipStream_t stream) {
  const float* cls_pred = (const float*)d_in[0];  // (32,80,128,128)
  const float* reg_pred = (const float*)d_in[1];  // (32,4,128,128)
  const float* obj_pred = (const float*)d_in[2];  // (32,1,128,128)
  const int*   tcls     = (const int*)d_in[3];    // (32,32)
  const float* tbox     = (const float*)d_in[4];  // (32,32,4)
  float* out = (float*)d_out;                     // 5 floats
  float* acc = (float*)d_ws;                      // 8 floats of scratch

  (void)in_sizes; (void)n_in; (void)out_size; (void)ws_size;

  yolo_init<<<1, 32, 0, stream>>>(acc);

  int n4 = (B_ * G_ * G_) / 4;  // 131072 float4's of obj_pred
  yolo_obj_bg<<<512, 256, 0, stream>>>(obj_pred, n4, acc);

  yolo_mask<<<B_, N_, 0, stream>>>(cls_pred, reg_pred, obj_pred, tcls, tbox, acc);

  yolo_finalize<<<1, 1, 0, stream>>>(acc, out);
}